// OptFlow_76038101008968
// MI455X (gfx1250) — compile-verified
//
#include <hip/hip_runtime.h>
#include <hip/hip_bf16.h>
#include <math.h>

typedef float v2f __attribute__((ext_vector_type(2)));
typedef float v8f __attribute__((ext_vector_type(8)));

#define BB   16            // batches
#define KC   4             // channels
#define HH   256
#define WW   256
#define NPB  (KC*HH*WW)    // 262144 pixels per batch
#define TOLF 0.001f
#define CHUNKS 8           // blocks per batch for hessian
#define HESS_BLOCK 256     // 8 waves
#define PIX_PER_THREAD 8
#define ACC_BLOCK 256
#define MAX_ITR_LAUNCH 20

// ---- workspace layout (floats) ----
// Hpad : 16*256   @ 0
// invH : 16*64    @ 4096
// p    : 16*8     @ 5120
// dp   : 16*8     @ 5248
// dpsum: 16*8     @ 5376

// ---- init: zero Hpad, p = init_param, dp = 1, dpsum = 0 ----
__global__ void of_init(const float* __restrict__ init_param, float* __restrict__ ws) {
    int i = blockIdx.x * 256 + threadIdx.x;
    if (i < 4096) ws[i] = 0.0f;                 // Hpad
    if (i < 128) {
        ws[5120 + i] = init_param[i];           // p
        ws[5248 + i] = 1.0f;                    // dp
        ws[5376 + i] = 0.0f;                    // dpsum
    }
}

// branchless J-column value: J[pix][par] = a(X,Y)*gx + b(X,Y)*gy with
// per-lane-constant polynomial weights (wa_*, wb_*).
__device__ __forceinline__ float jval_w(const float* __restrict__ tb, int pix,
                                        float wa_x, float wa_y, float wa_1,
                                        float wa_xx, float wa_xy,
                                        float wb_x, float wb_y, float wb_1,
                                        float wb_xy, float wb_yy) {
    int c   = pix >> 16;        // H*W = 65536
    int rem = pix & 65535;
    int y   = rem >> 8;
    int x   = rem & 255;
    const float* tc = tb + (c << 16);
    float tl = tc[y * WW + max(x - 1, 0)];
    float tr = tc[y * WW + min(x + 1, WW - 1)];
    float tu = tc[max(y - 1, 0) * WW + x];
    float td = tc[min(y + 1, HH - 1) * WW + x];
    float gx = 0.5f * (tr - tl);
    float gy = 0.5f * (td - tu);
    float X = (float)x - 127.5f;
    float Y = (float)y - 127.5f;
    float XX = X * X, XY = X * Y, YY = Y * Y;
    float a = wa_x * X + wa_y * Y + wa_1 + wa_xx * XX + wa_xy * XY;
    float b = wb_x * X + wb_y * Y + wb_1 + wb_xy * XY + wb_yy * YY;
    return a * gx + b * gy;
}

// ---- Hessian: Hm = J^T J per batch via V_WMMA_F32_16X16X4_F32 ----
// A (16x4): lane L<16 holds row M=L, v0/v1 = K0/K1; lane L>=16 holds K2/K3.
// B (4x16): lane L<16 holds col N=L, v0/v1 = K0/K1; lane L>=16 holds K2/K3.
// With B = J-chunk (K = pixel) these contents are identical -> pass `a` twice.
__global__ void of_hessian(const float* __restrict__ temp, float* __restrict__ ws) {
    float* Hpad = ws;
    const int b     = blockIdx.x / CHUNKS;
    const int chunk = blockIdx.x % CHUNKS;
    const int wave  = threadIdx.x >> 5;
    const int lane  = threadIdx.x & 31;
    const int par   = lane & 15;
    const int hi    = lane >> 4;                 // 0 -> pixels 0,1 ; 1 -> pixels 2,3
    const int wavesPerBatch = CHUNKS * (HESS_BLOCK / 32);   // 64
    const int waveId = chunk * (HESS_BLOCK / 32) + wave;
    const float* tb = temp + (size_t)b * NPB;

    // per-lane constant polynomial weights (zero for par >= 8)
    float wa_x  = (par == 0) ? 1.0f : 0.0f;
    float wa_y  = (par == 1) ? 1.0f : 0.0f;
    float wa_1  = (par == 2) ? 1.0f : 0.0f;
    float wb_x  = (par == 3) ? 1.0f : 0.0f;
    float wb_y  = (par == 4) ? 1.0f : 0.0f;
    float wb_1  = (par == 5) ? 1.0f : 0.0f;
    float wa_xx = (par == 6) ? -1.0f : 0.0f;
    float wb_xy = (par == 6) ? -1.0f : 0.0f;
    float wa_xy = (par == 7) ? -1.0f : 0.0f;
    float wb_yy = (par == 7) ? -1.0f : 0.0f;

    v8f acc = {};
    for (int base = waveId * 4; base < NPB; base += wavesPerBatch * 4) {
        int pA = base + hi * 2;
        v2f a;
        a.x = jval_w(tb, pA,     wa_x, wa_y, wa_1, wa_xx, wa_xy,
                                 wb_x, wb_y, wb_1, wb_xy, wb_yy);
        a.y = jval_w(tb, pA + 1, wa_x, wa_y, wa_1, wa_xx, wa_xy,
                                 wb_x, wb_y, wb_1, wb_xy, wb_yy);
        acc = __builtin_amdgcn_wmma_f32_16x16x4_f32(
            /*neg_a=*/false, a, /*neg_b=*/false, a,
            /*c_mod=*/(short)0, acc, /*reuse_a=*/false, /*reuse_b=*/false);
    }
    // C layout: VGPR v, lanes 0-15 -> (M=v, N=lane); lanes 16-31 -> (M=8+v, N=lane-16)
#pragma unroll
    for (int v = 0; v < 8; ++v) {
        atomicAdd(&Hpad[b * 256 + (hi * 8 + v) * 16 + par], acc[v]);
    }
}

// ---- invert 8x8 Hessians (Gauss-Jordan, one thread per batch) ----
__global__ void of_invert(float* __restrict__ ws) {
    int b = threadIdx.x;
    if (b >= BB) return;
    const float* Hpad = ws;
    float* invH = ws + 4096;
    float a[8][8], iv[8][8];
    for (int i = 0; i < 8; ++i)
        for (int j = 0; j < 8; ++j) {
            a[i][j]  = Hpad[b * 256 + i * 16 + j];
            iv[i][j] = (i == j) ? 1.0f : 0.0f;
        }
    for (int col = 0; col < 8; ++col) {
        int piv = col; float best = fabsf(a[col][col]);
        for (int r = col + 1; r < 8; ++r) {
            float v = fabsf(a[r][col]);
            if (v > best) { best = v; piv = r; }
        }
        if (piv != col) {
            for (int j = 0; j < 8; ++j) {
                float t = a[col][j]; a[col][j] = a[piv][j]; a[piv][j] = t;
                t = iv[col][j]; iv[col][j] = iv[piv][j]; iv[piv][j] = t;
            }
        }
        float d = 1.0f / a[col][col];
        for (int j = 0; j < 8; ++j) { a[col][j] *= d; iv[col][j] *= d; }
        for (int r = 0; r < 8; ++r) {
            if (r == col) continue;
            float f = a[r][col];
            for (int j = 0; j < 8; ++j) {
                a[r][j]  -= f * a[col][j];
                iv[r][j] -= f * iv[col][j];
            }
        }
    }
    for (int i = 0; i < 8; ++i)
        for (int j = 0; j < 8; ++j)
            invH[b * 64 + i * 8 + j] = iv[i][j];
}

// ---- per-iteration: warp + residual + J^T r accumulation ----
__global__ void of_accum(const float* __restrict__ img, const float* __restrict__ temp,
                         float* __restrict__ ws, const int* __restrict__ max_itr, int iter) {
    if (iter >= *max_itr) return;
    const float* p  = ws + 5120;
    float* dpsum    = ws + 5376;
    const int tid = threadIdx.x;
    const size_t blockBase = (size_t)blockIdx.x * (ACC_BLOCK * PIX_PER_THREAD);
    const int b = (int)(blockBase / NPB);

    float p0 = p[b*8+0], p1 = p[b*8+1], p2 = p[b*8+2], p3 = p[b*8+3];
    float p4 = p[b*8+4], p5 = p[b*8+5], p6 = p[b*8+6], p7 = p[b*8+7];
    float h00 = 1.0f + p0, h01 = p1, h02 = p2;
    float h10 = p3, h11 = 1.0f + p4, h12 = p5;
    float h20 = p6, h21 = p7;

    const float* ib0 = img  + (size_t)b * NPB;
    const float* tb0 = temp + (size_t)b * NPB;
    const float bx = 1.0f - 2.0f / (float)WW;
    const float by = 1.0f - 2.0f / (float)HH;

    float aj0=0, aj1=0, aj2=0, aj3=0, aj4=0, aj5=0, aj6=0, aj7=0;

#pragma unroll
    for (int s = 0; s < PIX_PER_THREAD; ++s) {
        int local = (int)(blockBase - (size_t)b * NPB) + s * ACC_BLOCK + tid;
        int c   = local >> 16;
        int rem = local & 65535;
        int y   = rem >> 8;
        int x   = rem & 255;
        float X = (float)x - 127.5f;
        float Y = (float)y - 127.5f;

        float xw = h00 * X + h01 * Y + h02;
        float yw = h10 * X + h11 * Y + h12;
        float zw = h20 * X + h21 * Y + 1.0f;
        float iz = 1.0f / zw;
        float Xw = xw * iz + 127.5f;
        float Yw = yw * iz + 127.5f;
        float x0f = floorf(Xw), y0f = floorf(Yw);
        float wx = Xw - x0f, wy = Yw - y0f;
        int x0 = (int)x0f, y0 = (int)y0f;

        const float* ic = ib0 + (c << 16);
        float v00, v01, v10, v11;
        {
            int xc, yc; bool vv;
            vv = (x0 >= 0) && (x0 < WW) && (y0 >= 0) && (y0 < HH);
            xc = min(max(x0, 0), WW-1); yc = min(max(y0, 0), HH-1);
            v00 = vv ? ic[yc * WW + xc] : 0.0f;
            vv = (x0+1 >= 0) && (x0+1 < WW) && (y0 >= 0) && (y0 < HH);
            xc = min(max(x0+1, 0), WW-1);
            v01 = vv ? ic[yc * WW + xc] : 0.0f;
            vv = (x0 >= 0) && (x0 < WW) && (y0+1 >= 0) && (y0+1 < HH);
            xc = min(max(x0, 0), WW-1); yc = min(max(y0+1, 0), HH-1);
            v10 = vv ? ic[yc * WW + xc] : 0.0f;
            vv = (x0+1 >= 0) && (x0+1 < WW) && (y0+1 >= 0) && (y0+1 < HH);
            xc = min(max(x0+1, 0), WW-1);
            v11 = vv ? ic[yc * WW + xc] : 0.0f;
        }
        float Q = v00 * (1.0f-wx) * (1.0f-wy) + v01 * wx * (1.0f-wy)
                + v10 * (1.0f-wx) * wy        + v11 * wx * wy;

        float xn = Xw * (1.0f / 127.5f) - 1.0f;
        float yn = Yw * (1.0f / 127.5f) - 1.0f;
        float mask = ((xn > -bx) && (xn < bx) && (yn > -by) && (yn < by)) ? 1.0f : 0.0f;

        const float* tc = tb0 + (c << 16);
        float r = Q - tc[y * WW + x] * mask;

        float tl = tc[y * WW + max(x - 1, 0)];
        float tr = tc[y * WW + min(x + 1, WW - 1)];
        float tu = tc[max(y - 1, 0) * WW + x];
        float td = tc[min(y + 1, HH - 1) * WW + x];
        float gx = 0.5f * (tr - tl);
        float gy = 0.5f * (td - tu);

        aj0 += X * gx * r;
        aj1 += Y * gx * r;
        aj2 += gx * r;
        aj3 += X * gy * r;
        aj4 += Y * gy * r;
        aj5 += gy * r;
        aj6 += (-X * X * gx - X * Y * gy) * r;
        aj7 += (-X * Y * gx - Y * Y * gy) * r;
    }

    __shared__ float sred[8];
    if (tid < 8) sred[tid] = 0.0f;
    __syncthreads();
    float accj[8] = {aj0, aj1, aj2, aj3, aj4, aj5, aj6, aj7};
#pragma unroll
    for (int j = 0; j < 8; ++j) {
        float v = accj[j];
        for (int o = 16; o > 0; o >>= 1) v += __shfl_down(v, o, 32);
        if ((tid & 31) == 0) atomicAdd(&sred[j], v);
    }
    __syncthreads();
    if (tid < 8) atomicAdd(&dpsum[b * 8 + tid], sred[tid]);
}

// ---- per-iteration: dp = gate * (invH @ dpsum), gated p update, re-zero dpsum ----
__global__ void of_update(float* __restrict__ ws, const int* __restrict__ max_itr, int iter) {
    if (iter >= *max_itr) return;
    const float* invH = ws + 4096;
    float* p     = ws + 5120;
    float* dp    = ws + 5248;
    float* dpsum = ws + 5376;
    int tid = threadIdx.x;              // 128 threads
    int b = tid >> 3, r = tid & 7;

    __shared__ float nrm2[BB];
    __shared__ float sdpraw[128];
    __shared__ float smax;
    float dprev = dp[tid];
    if (r == 0) nrm2[b] = 0.0f;
    __syncthreads();
    atomicAdd(&nrm2[b], dprev * dprev);
    sdpraw[tid] = dpsum[tid];
    __syncthreads();
    if (tid == 0) {
        float m = 0.0f;
        for (int i = 0; i < BB; ++i) m = fmaxf(m, sqrtf(nrm2[i]));
        smax = m;
    }
    __syncthreads();

    bool  cond = (smax > TOLF) || (iter == 0);
    float gate = (sqrtf(nrm2[b]) > TOLF) ? 1.0f : 0.0f;

    float dnew = 0.0f;
#pragma unroll
    for (int j = 0; j < 8; ++j) dnew += invH[b * 64 + r * 8 + j] * sdpraw[b * 8 + j];
    if (r >= 6) dnew = 0.0f;
    dnew *= gate;

    if (cond) {
        p[tid]  = p[tid] - dnew;
        dp[tid] = dnew;
    }
    dpsum[tid] = 0.0f;
}

// ---- write outputs: p (16,8,1) then H = I + reshape([p,0]) (16,3,3) ----
__global__ void of_final(const float* __restrict__ ws, float* __restrict__ out) {
    const float* p = ws + 5120;
    int i = threadIdx.x;                // 288 threads
    if (i < 128) {
        out[i] = p[i];
    } else if (i < 272) {
        int t = i - 128;
        int b = t / 9, e = t % 9;
        float v = (e < 8) ? p[b * 8 + e] : 0.0f;
        out[i] = v + ((e / 3 == e % 3) ? 1.0f : 0.0f);
    }
}

extern "C" void kernel_launch(void* const* d_in, const int* in_sizes, int n_in,
                              void* d_out, int out_size, void* d_ws, size_t ws_size,
                              hipStream_t stream) {
    const float* img        = (const float*)d_in[0];
    const float* temp       = (const float*)d_in[1];
    const float* init_param = (const float*)d_in[2];
    const int*   max_itr    = (const int*)d_in[3];
    float* out = (float*)d_out;
    float* ws  = (float*)d_ws;

    of_init<<<16, 256, 0, stream>>>(init_param, ws);
    of_hessian<<<BB * CHUNKS, HESS_BLOCK, 0, stream>>>(temp, ws);
    of_invert<<<1, 32, 0, stream>>>(ws);

    const int accBlocks = (BB * NPB) / (ACC_BLOCK * PIX_PER_THREAD);  // 2048
    for (int i = 0; i < MAX_ITR_LAUNCH; ++i) {
        of_accum<<<accBlocks, ACC_BLOCK, 0, stream>>>(img, temp, ws, max_itr, i);
        of_update<<<1, 128, 0, stream>>>(ws, max_itr, i);
    }
    of_final<<<1, 288, 0, stream>>>(ws, out);
}